// GCNEncoder_13975823581699
// MI455X (gfx1250) — compile-verified
//
#include <hip/hip_runtime.h>

typedef __attribute__((ext_vector_type(2))) float v2f;
typedef __attribute__((ext_vector_type(8))) float v8f;

#define IN_C 128
#define HID  32
#define LAT  16

// ---------- degree / normalization ----------
__global__ __launch_bounds__(256) void k_init_deg(float* __restrict__ deg, int n) {
    int i = blockIdx.x * 256 + threadIdx.x;
    if (i < n) deg[i] = 1.0f;               // self-loop contributes 1
}

__global__ __launch_bounds__(256) void k_deg(const long long* __restrict__ dst,
                                             float* __restrict__ deg, int E) {
    int e = blockIdx.x * 256 + threadIdx.x;
    if (e < E) unsafeAtomicAdd(&deg[dst[e]], 1.0f);
}

__global__ __launch_bounds__(256) void k_rsqrt(const float* __restrict__ deg,
                                               float* __restrict__ dis, int n) {
    int i = blockIdx.x * 256 + threadIdx.x;
    if (i < n) {
        float d = deg[i];
        dis[i] = d > 0.0f ? rsqrtf(d) : 0.0f;
    }
}

// ---------- layer 1 dense transform: tt = (x @ W1) * dis, agg1 init = tt * dis ----------
// One wave computes a 16x32 output tile with v_wmma_f32_16x16x4_f32, K=128.
// W1 is staged once per block into LDS, pre-swizzled into B-fragment order so
// each k-step B operand is a single conflict-free ds_load_b64 per lane.
__global__ __launch_bounds__(256) void k_gemm1(const float* __restrict__ x,
                                               const float* __restrict__ W1,
                                               const float* __restrict__ dis,
                                               float* __restrict__ tt,
                                               float* __restrict__ agg1, int n) {
    // w1f[kb*2 + tile][lane] = { W1[kb*4 + khalf*2 + 0][col], W1[kb*4 + khalf*2 + 1][col] }
    //   khalf = lane>>4, col = (lane&15) + tile*16
    __shared__ float2 w1f[(IN_C / 4) * 2][32];          // 2048 * 8B = 16 KB

    // ---- cooperative pre-swizzled staging (all 256 threads) ----
    for (int idx = threadIdx.x; idx < (IN_C / 4) * 2 * 32; idx += 256) {
        int lane_i = idx & 31;
        int fi     = idx >> 5;                 // 0..63  = kb*2 + tile
        int tile   = fi & 1;
        int kb     = fi >> 1;
        int khalf  = lane_i >> 4;
        int col    = (lane_i & 15) + tile * 16;
        int brow   = kb * 4 + khalf * 2;
        float2 v;
        v.x = W1[brow * HID + col];
        v.y = W1[(brow + 1) * HID + col];
        w1f[fi][lane_i] = v;
    }
    __syncthreads();

    int wid  = blockIdx.x * 8 + (threadIdx.x >> 5);
    int lane = threadIdx.x & 31;
    int row0 = wid * 16;
    if (row0 >= n) return;                    // wave-uniform (after barrier)
    int mrow  = lane & 15;                    // tile row / B column
    int khalf = lane >> 4;                    // 0: K={0,1}, 1: K={2,3}

    int arow_i = row0 + mrow;
    if (arow_i >= n) arow_i = n - 1;          // clamp (n is mult of 16, safety only)
    const float* arow = x + (size_t)arow_i * IN_C + khalf * 2;

    v8f acc0 = {}; v8f acc1 = {};
    for (int kb = 0; kb < IN_C / 4; ++kb) {
        v2f a; a.x = arow[kb * 4]; a.y = arow[kb * 4 + 1];   // A[16x4] fragment (b64)
        float2 f0 = w1f[kb * 2 + 0][lane];                   // B cols 0..15  (ds b64)
        float2 f1 = w1f[kb * 2 + 1][lane];                   // B cols 16..31 (ds b64)
        v2f b0; b0.x = f0.x; b0.y = f0.y;
        v2f b1; b1.x = f1.x; b1.y = f1.y;
        acc0 = __builtin_amdgcn_wmma_f32_16x16x4_f32(false, a, false, b0,
                                                     (short)0, acc0, false, false);
        acc1 = __builtin_amdgcn_wmma_f32_16x16x4_f32(false, a, false, b1,
                                                     (short)0, acc1, false, false);
    }
    // D layout: VGPR r -> row r (lanes 0-15) / row r+8 (lanes 16-31), col = lane&15
    int rbase = row0 + (khalf ? 8 : 0);
    int col   = mrow;
    for (int r = 0; r < 8; ++r) {
        int rr = rbase + r;
        if (rr >= n) break;
        float di = dis[rr];
        float t0 = acc0[r] * di;
        float t1 = acc1[r] * di;
        size_t o = (size_t)rr * HID + col;
        tt[o]        = t0;        tt[o + 16]   = t1;      // gathered by edges
        agg1[o]      = t0 * di;   agg1[o + 16] = t1 * di; // self-loop init
    }
}

// ---------- layer 1 edge scatter: agg1[d] += tt[s] * dis[d] ----------
__global__ __launch_bounds__(256) void k_scatter1(const long long* __restrict__ src,
                                                  const long long* __restrict__ dst,
                                                  const float* __restrict__ dis,
                                                  const float* __restrict__ tt,
                                                  float* __restrict__ agg1, int E) {
    int tid = blockIdx.x * 256 + threadIdx.x;
    int e = tid >> 3;                         // 8 threads per edge (32 ch / 4)
    if (e >= E) return;
    int q = tid & 7;
    long long s = src[e], d = dst[e];
    float w = dis[d];
    float4 v = *((const float4*)(tt + (size_t)s * HID) + q);
    float* op = agg1 + (size_t)d * HID + q * 4;
    unsafeAtomicAdd(op + 0, v.x * w);
    unsafeAtomicAdd(op + 1, v.y * w);
    unsafeAtomicAdd(op + 2, v.z * w);
    unsafeAtomicAdd(op + 3, v.w * w);
}

// ---------- layer 2: h = relu(agg1 + b1); uu = (h @ W2) * dis; out init = uu*dis + b2 ----------
__global__ __launch_bounds__(256) void k_gemm2(const float* __restrict__ agg1,
                                               const float* __restrict__ b1,
                                               const float* __restrict__ W2,
                                               const float* __restrict__ b2,
                                               const float* __restrict__ dis,
                                               float* __restrict__ uu,
                                               float* __restrict__ out, int n) {
    // W2 fragments: [kb][lane] = { W2[k0][col], W2[k0+1][col] }, k0 = kb*4 + khalf*2
    __shared__ float2 w2f[HID / 4][32];       // 256 * 8B = 2 KB
    __shared__ float2 b1f[HID / 4][2];        // bias pairs per (kb, khalf)

    for (int idx = threadIdx.x; idx < (HID / 4) * 32; idx += 256) {
        int lane_i = idx & 31;
        int kb     = idx >> 5;
        int khalf  = lane_i >> 4;
        int col    = lane_i & 15;
        int k0     = kb * 4 + khalf * 2;
        float2 v;
        v.x = W2[k0 * LAT + col];
        v.y = W2[(k0 + 1) * LAT + col];
        w2f[kb][lane_i] = v;
    }
    if (threadIdx.x < (HID / 4) * 2) {
        int kb = threadIdx.x >> 1, kh = threadIdx.x & 1;
        int k0 = kb * 4 + kh * 2;
        float2 v; v.x = b1[k0]; v.y = b1[k0 + 1];
        b1f[kb][kh] = v;
    }
    __syncthreads();

    int wid  = blockIdx.x * 8 + (threadIdx.x >> 5);
    int lane = threadIdx.x & 31;
    int row0 = wid * 16;
    if (row0 >= n) return;
    int mrow  = lane & 15;
    int khalf = lane >> 4;

    int arow_i = row0 + mrow;
    if (arow_i >= n) arow_i = n - 1;
    const float* arow = agg1 + (size_t)arow_i * HID + khalf * 2;

    v8f acc = {};
    for (int kb = 0; kb < HID / 4; ++kb) {
        float2 bf = b1f[kb][khalf];
        v2f a;                                             // fused bias + ReLU
        a.x = fmaxf(arow[kb * 4]     + bf.x, 0.0f);
        a.y = fmaxf(arow[kb * 4 + 1] + bf.y, 0.0f);
        float2 f = w2f[kb][lane];
        v2f b; b.x = f.x; b.y = f.y;
        acc = __builtin_amdgcn_wmma_f32_16x16x4_f32(false, a, false, b,
                                                    (short)0, acc, false, false);
    }
    int rbase = row0 + (khalf ? 8 : 0);
    int col   = mrow;
    float bc  = b2[col];
    for (int r = 0; r < 8; ++r) {
        int rr = rbase + r;
        if (rr >= n) break;
        float di = dis[rr];
        float u  = acc[r] * di;                 // uu = h@W2 * dis[s]
        size_t o = (size_t)rr * LAT + col;
        uu[o]  = u;
        out[o] = u * di + bc;                   // self-loop + bias
    }
}

// ---------- layer 2 edge scatter: out[d] += uu[s] * dis[d] ----------
__global__ __launch_bounds__(256) void k_scatter2(const long long* __restrict__ src,
                                                  const long long* __restrict__ dst,
                                                  const float* __restrict__ dis,
                                                  const float* __restrict__ uu,
                                                  float* __restrict__ out, int E) {
    int tid = blockIdx.x * 256 + threadIdx.x;
    int e = tid >> 2;                         // 4 threads per edge (16 ch / 4)
    if (e >= E) return;
    int q = tid & 3;
    long long s = src[e], d = dst[e];
    float w = dis[d];
    float4 v = *((const float4*)(uu + (size_t)s * LAT) + q);
    float* op = out + (size_t)d * LAT + q * 4;
    unsafeAtomicAdd(op + 0, v.x * w);
    unsafeAtomicAdd(op + 1, v.y * w);
    unsafeAtomicAdd(op + 2, v.z * w);
    unsafeAtomicAdd(op + 3, v.w * w);
}

extern "C" void kernel_launch(void* const* d_in, const int* in_sizes, int n_in,
                              void* d_out, int out_size, void* d_ws, size_t ws_size,
                              hipStream_t stream) {
    const float*     x  = (const float*)d_in[0];
    const long long* ei = (const long long*)d_in[1];   // int64 edge_index [2,E]
    const float*     W1 = (const float*)d_in[2];
    const float*     b1 = (const float*)d_in[3];
    const float*     W2 = (const float*)d_in[4];
    const float*     b2 = (const float*)d_in[5];

    int n = in_sizes[0] / IN_C;
    int E = in_sizes[1] / 2;
    const long long* src = ei;
    const long long* dst = ei + E;

    float* ws   = (float*)d_ws;
    float* deg  = ws;                          // n
    float* dis  = deg  + n;                    // n
    float* tt   = dis  + n;                    // n*32
    float* agg1 = tt   + (size_t)n * HID;      // n*32
    float* uu   = agg1 + (size_t)n * HID;      // n*16
    float* out  = (float*)d_out;               // n*16

    int nb = (n + 255) / 256;
    int nwaves = (n + 15) / 16;
    int gb = (nwaves + 7) / 8;

    k_init_deg<<<nb, 256, 0, stream>>>(deg, n);
    k_deg<<<(E + 255) / 256, 256, 0, stream>>>(dst, deg, E);
    k_rsqrt<<<nb, 256, 0, stream>>>(deg, dis, n);
    k_gemm1<<<gb, 256, 0, stream>>>(x, W1, dis, tt, agg1, n);
    k_scatter1<<<(E * 8 + 255) / 256, 256, 0, stream>>>(src, dst, dis, tt, agg1, E);
    k_gemm2<<<gb, 256, 0, stream>>>(agg1, b1, W2, b2, dis, uu, out, n);
    k_scatter2<<<(E * 4 + 255) / 256, 256, 0, stream>>>(src, dst, dis, uu, out, E);
}